// SAGEConv_54554674594336
// MI455X (gfx1250) — compile-verified
//
#include <hip/hip_runtime.h>
#include <hip/hip_bf16.h>
#include <stdint.h>

typedef __attribute__((ext_vector_type(2))) float v2f;
typedef __attribute__((ext_vector_type(8))) float v8f;

#define DIM 128
#define TWO_D 256

// ---------------- degree counting ----------------
__global__ void k_zero_u32(unsigned* __restrict__ p, int n) {
    int i = blockIdx.x * blockDim.x + threadIdx.x;
    if (i < n) p[i] = 0u;
}

__global__ void k_count(const int* __restrict__ senders,
                        const int* __restrict__ receivers,
                        unsigned* __restrict__ cntOut,   // over receivers
                        unsigned* __restrict__ cntIn,    // over senders
                        int E) {
    int e = blockIdx.x * blockDim.x + threadIdx.x;
    if (e < E) {
        atomicAdd(&cntOut[receivers[e]], 1u);
        atomicAdd(&cntIn[senders[e]], 1u);
    }
}

// ---------------- per-node normalization + self-loop init ----------------
// out_degree = cntOut+1 (self loop), in_degree = cntIn+1.
// outNorm = rsqrt(out_deg); scaleIn = rsqrt(in_deg)/out_deg  (in_norm / edge_count)
// agg[n]  = x[n] * outNorm[n]   (the self-loop contribution; also clears poison)
__global__ void k_node_prep(const float* __restrict__ x,
                            const unsigned* __restrict__ cntOut,
                            const unsigned* __restrict__ cntIn,
                            float* __restrict__ outNorm,
                            float* __restrict__ scaleIn,
                            float* __restrict__ agg, int N) {
    int n = blockIdx.x;            // one block (128 threads) per node
    __shared__ float sNorm;
    if (threadIdx.x == 0) {
        float od = (float)(cntOut[n] + 1u);
        float id = (float)(cntIn[n] + 1u);
        float on = rsqrtf(od);
        outNorm[n] = on;
        scaleIn[n] = rsqrtf(id) / od;
        sNorm = on;
    }
    __syncthreads();
    size_t o = (size_t)n * DIM + threadIdx.x;
    agg[o] = x[o] * sNorm;
}

// ---------------- edge scatter: one wave32 per edge, float4 per lane ----------------
__global__ void k_scatter(const float* __restrict__ x,
                          const int* __restrict__ senders,
                          const int* __restrict__ receivers,
                          const float* __restrict__ outNorm,
                          float* __restrict__ agg, int E) {
    int e    = blockIdx.x * (blockDim.x >> 5) + (threadIdx.x >> 5);
    int lane = threadIdx.x & 31;
    if (e >= E) return;
    int s = senders[e];
    int r = receivers[e];
    float w = outNorm[s];
    const float4 v = *(const float4*)(x + (size_t)s * DIM + lane * 4);
    float* dst = agg + (size_t)r * DIM + lane * 4;
    atomicAdd(dst + 0, v.x * w);
    atomicAdd(dst + 1, v.y * w);
    atomicAdd(dst + 2, v.z * w);
    atomicAdd(dst + 3, v.w * w);
}

// ---------------- fused scale + GEMM: out = [x | agg*scaleIn] @ W^T + b ----------------
// 2 waves per block; each wave computes a 16(M=nodes) x 128(N=out dims) strip via
// v_wmma_f32_16x16x4_f32 over K = 256 in steps of 4 (8 N-tiles x 64 k-steps).
#define A_PITCH 260   // floats; 1040B rows: 16B-aligned, bank-spread
__global__ __launch_bounds__(64) void k_gemm(
        const float* __restrict__ x, const float* __restrict__ agg,
        const float* __restrict__ scaleIn, const float* __restrict__ W,
        const float* __restrict__ bias, float* __restrict__ out, int N) {
    __shared__ float sA[2][16 * A_PITCH];   // per-wave 16 x 256 combined rows

    const int wave = threadIdx.x >> 5;
    const int lane = threadIdx.x & 31;
    const int half = lane >> 4;             // 0: lanes 0-15, 1: lanes 16-31
    const int lan  = lane & 15;
    const int tileBase = (blockIdx.x * 2 + wave) * 16;

    // --- stage A tile: cols [0,128) = x, cols [128,256) = agg * scaleIn ---
    float* sAw = sA[wave];
    for (int r = 0; r < 16; ++r) {
        int row = tileBase + r;
        if (row >= N) row = N - 1;           // clamp; duplicate rows never stored
        const float4 xv = *(const float4*)(x   + (size_t)row * DIM + lane * 4);
        const float4 av = *(const float4*)(agg + (size_t)row * DIM + lane * 4);
        float sc = scaleIn[row];
        float4 as = make_float4(av.x * sc, av.y * sc, av.z * sc, av.w * sc);
        *(float4*)(sAw + r * A_PITCH +       lane * 4) = xv;
        *(float4*)(sAw + r * A_PITCH + DIM + lane * 4) = as;
    }
    __syncthreads();

    // --- accumulators: 8 tiles of 16x16, bias folded in (bias depends on col only) ---
    v8f acc[8];
#pragma unroll
    for (int t = 0; t < 8; ++t) {
        float bv = bias[t * 16 + lan];
        acc[t] = (v8f){bv, bv, bv, bv, bv, bv, bv, bv};
    }

    // A fragment layout (16x4 f32): lanes 0-15 -> K{k0,k0+1}, lanes 16-31 -> K{k0+2,k0+3}
    const int kHi = half * 2;
    for (int k0 = 0; k0 < TWO_D; k0 += 4) {
        const int kk = k0 + kHi;
        v2f a = *(const v2f*)(sAw + lan * A_PITCH + kk);   // ds_load_b64
#pragma unroll
        for (int t = 0; t < 8; ++t) {
            // B fragment (4x16): B[k][n] = W[n][k]; same K split as A, N = lan
            v2f bf = *(const v2f*)(W + (size_t)(t * 16 + lan) * TWO_D + kk);
            acc[t] = __builtin_amdgcn_wmma_f32_16x16x4_f32(
                false, a, false, bf, (short)0, acc[t], false, false);
        }
    }

    // --- store: C/D layout: VGPR j -> row j (lanes 0-15) / row j+8 (lanes 16-31) ---
#pragma unroll
    for (int t = 0; t < 8; ++t) {
#pragma unroll
        for (int j = 0; j < 8; ++j) {
            int row = tileBase + j + half * 8;
            if (row < N) out[(size_t)row * DIM + t * 16 + lan] = acc[t][j];
        }
    }
}

extern "C" void kernel_launch(void* const* d_in, const int* in_sizes, int n_in,
                              void* d_out, int out_size, void* d_ws, size_t ws_size,
                              hipStream_t stream) {
    const float* x        = (const float*)d_in[0];
    const int*   senders  = (const int*)d_in[1];
    const int*   receivers= (const int*)d_in[2];
    // d_in[3] = n_nodes scalar (derived from in_sizes instead)
    const float* W        = (const float*)d_in[4];
    const float* bias     = (const float*)d_in[5];
    float*       out      = (float*)d_out;

    const int N = in_sizes[0] / DIM;
    const int E = in_sizes[1];

    // workspace layout (~26.4 MB): agg first so it is 16B-aligned for float4
    float*    agg     = (float*)d_ws;                  // N*DIM
    float*    outNorm = agg + (size_t)N * DIM;         // N
    float*    scaleIn = outNorm + N;                   // N
    unsigned* cntOut  = (unsigned*)(scaleIn + N);      // N
    // cntIn = cntOut + N (contiguous, zeroed together)
    unsigned* cntIn   = cntOut + N;                    // N

    k_zero_u32<<<(2 * N + 255) / 256, 256, 0, stream>>>(cntOut, 2 * N);
    k_count   <<<(E + 255) / 256, 256, 0, stream>>>(senders, receivers, cntOut, cntIn, E);
    k_node_prep<<<N, DIM, 0, stream>>>(x, cntOut, cntIn, outNorm, scaleIn, agg, N);
    k_scatter <<<(E + 7) / 8, 256, 0, stream>>>(x, senders, receivers, outNorm, agg, E);
    k_gemm    <<<(N + 31) / 32, 64, 0, stream>>>(x, agg, scaleIn, W, bias, out, N);
}